// RNN_45561013076393
// MI455X (gfx1250) — compile-verified
//
#include <hip/hip_runtime.h>

// Probe results: __builtin_amdgcn_tanhf PRESENT, global_load_async_to_lds_b32/b128
// PRESENT (b128: (AS1 int4*, AS3 ptr, imm, imm)), s_wait_asynccnt PRESENT.

#if defined(__HIP_DEVICE_COMPILE__) && \
    __has_builtin(__builtin_amdgcn_global_load_async_to_lds_b128) && \
    __has_builtin(__builtin_amdgcn_s_wait_asynccnt)
#define HAVE_ASYNC_DEV 1
#else
#define HAVE_ASYNC_DEV 0
#endif

#if defined(__HIP_DEVICE_COMPILE__) && \
    __has_builtin(__builtin_amdgcn_wmma_f32_16x16x4_f32)
#define HAVE_WMMA_F32X4 1
#else
#define HAVE_WMMA_F32X4 0
#endif

typedef float v2f __attribute__((ext_vector_type(2)));
typedef float v8f __attribute__((ext_vector_type(8)));

static __device__ __forceinline__ v2f make2(float a, float b) {
    v2f r; r.x = a; r.y = b; return r;
}
static __device__ __forceinline__ v2f splat2(float s) { return make2(s, s); }
// -> v_pk_fma_f32 (VOP3P packed fp32) on gfx1250
static __device__ __forceinline__ v2f pk_fma(v2f a, v2f b, v2f c) {
    return __builtin_elementwise_fma(a, b, c);
}

// ---------------- fast activations (native v_tanh_f32 trans op) ----------------
static __device__ __forceinline__ float fast_tanh(float x) {
#if __has_builtin(__builtin_amdgcn_tanhf)
    return __builtin_amdgcn_tanhf(x);
#else
    float e = __builtin_amdgcn_exp2f(x * -2.885390082f);   // exp(-2x)
    return __builtin_fmaf(2.0f, __builtin_amdgcn_rcpf(1.0f + e), -1.0f);
#endif
}
static __device__ __forceinline__ v2f tanh2(v2f v) {
    return make2(fast_tanh(v.x), fast_tanh(v.y));
}
static __device__ __forceinline__ float fast_sig(float x) {
#if __has_builtin(__builtin_amdgcn_tanhf)
    return __builtin_fmaf(0.5f, __builtin_amdgcn_tanhf(0.5f * x), 0.5f);
#else
    float e = __builtin_amdgcn_exp2f(x * -1.442695041f);
    return __builtin_amdgcn_rcpf(1.0f + e);
#endif
}

// ---------------- packed fused 2-layer LSTM cell, one time step (H=2, I=1) ----------
// Gate pairs p: 0=i, 1=f, 2=g(candidate), 3=o (PyTorch order). i/f/o weights are
// PRE-SCALED by 0.5 at setup so sigmoid(g) == fma(0.5, tanh(g'), 0.5) with no pre-mul.
static __device__ __forceinline__ void lstm2_step(
    float xv,
    const v2f* __restrict__ wih0,                                  // [4]
    const v2f* __restrict__ whh0a, const v2f* __restrict__ whh0b,  // col0/col1 pairs
    const v2f* __restrict__ bb0,                                   // [4]
    const v2f* __restrict__ wih1a, const v2f* __restrict__ wih1b,
    const v2f* __restrict__ whh1a, const v2f* __restrict__ whh1b,
    const v2f* __restrict__ bb1,
    v2f& h0, v2f& c0, v2f& h1, v2f& c1)
{
    const v2f half = splat2(0.5f);
    // ---- layer 0
    const v2f xs = splat2(xv);
    const v2f ha = splat2(h0.x), hb = splat2(h0.y);
    v2f gi = pk_fma(xs, wih0[0], pk_fma(ha, whh0a[0], pk_fma(hb, whh0b[0], bb0[0])));
    v2f gf = pk_fma(xs, wih0[1], pk_fma(ha, whh0a[1], pk_fma(hb, whh0b[1], bb0[1])));
    v2f gn = pk_fma(xs, wih0[2], pk_fma(ha, whh0a[2], pk_fma(hb, whh0b[2], bb0[2])));
    v2f go = pk_fma(xs, wih0[3], pk_fma(ha, whh0a[3], pk_fma(hb, whh0b[3], bb0[3])));
    v2f ai = pk_fma(half, tanh2(gi), half);
    v2f af = pk_fma(half, tanh2(gf), half);
    v2f an = tanh2(gn);
    v2f ao = pk_fma(half, tanh2(go), half);
    c0 = pk_fma(af, c0, ai * an);
    h0 = ao * tanh2(c0);
    // ---- layer 1 (input = layer-0 hidden pair)
    const v2f pa = splat2(h0.x), pb = splat2(h0.y);
    const v2f qa = splat2(h1.x), qb = splat2(h1.y);
    gi = pk_fma(pa, wih1a[0], pk_fma(pb, wih1b[0], pk_fma(qa, whh1a[0], pk_fma(qb, whh1b[0], bb1[0]))));
    gf = pk_fma(pa, wih1a[1], pk_fma(pb, wih1b[1], pk_fma(qa, whh1a[1], pk_fma(qb, whh1b[1], bb1[1]))));
    gn = pk_fma(pa, wih1a[2], pk_fma(pb, wih1b[2], pk_fma(qa, whh1a[2], pk_fma(qb, whh1b[2], bb1[2]))));
    go = pk_fma(pa, wih1a[3], pk_fma(pb, wih1b[3], pk_fma(qa, whh1a[3], pk_fma(qb, whh1b[3], bb1[3]))));
    ai = pk_fma(half, tanh2(gi), half);
    af = pk_fma(half, tanh2(gf), half);
    an = tanh2(gn);
    ao = pk_fma(half, tanh2(go), half);
    c1 = pk_fma(af, c1, ai * an);
    h1 = ao * tanh2(c1);
}

#if HAVE_ASYNC_DEV
// one async b128: copies 16B per lane, memory -> LDS, tracked by ASYNCcnt.
typedef int v4i __attribute__((__vector_size__(4 * sizeof(int))));
typedef __attribute__((address_space(1))) v4i* gv4i_p;
typedef __attribute__((address_space(3))) v4i* lv4i_p;

static __device__ __forceinline__ void async_ld16(const float* g, const float* l) {
    __builtin_amdgcn_global_load_async_to_lds_b128(
        (gv4i_p)(unsigned long long)g,
        (lv4i_p)(unsigned int)(unsigned long long)l,
        0, 0);
}
#endif

#define LSTM_CH   32   // timesteps staged per chunk
#define LDS_PITCH 36   // floats per LDS row (32 + 4 pad: 16B-aligned, de-conflicted)

// ---------------- kernel: one batch element per lane, full T scan ----------------
template <bool STAGE>
__global__ __launch_bounds__(32)
void RNN_45561013076393_kernel(
    const float* __restrict__ x,
    const float* __restrict__ W_ih_l0, const float* __restrict__ W_hh_l0,
    const float* __restrict__ b_ih_l0, const float* __restrict__ b_hh_l0,
    const float* __restrict__ W_ih_l1, const float* __restrict__ W_hh_l1,
    const float* __restrict__ b_ih_l1, const float* __restrict__ b_hh_l1,
    const float* __restrict__ W_dense, const float* __restrict__ b_dense,
    float* __restrict__ out, int B, int T)
{
    const int lane = threadIdx.x;
    const int b = blockIdx.x * 32 + lane;

    // Uniform weight prep (scalar-float SALU + SGPRs). Prescale i/f/o by 0.5.
    v2f wih0[4], whh0a[4], whh0b[4], bb0[4];
    v2f wih1a[4], wih1b[4], whh1a[4], whh1b[4], bb1[4];
#pragma unroll
    for (int p = 0; p < 4; ++p) {
        const int r0 = 2 * p, r1 = 2 * p + 1;
        const float s = (p == 2) ? 1.0f : 0.5f;
        wih0[p]  = make2(s * W_ih_l0[r0],          s * W_ih_l0[r1]);
        whh0a[p] = make2(s * W_hh_l0[2 * r0 + 0],  s * W_hh_l0[2 * r1 + 0]);
        whh0b[p] = make2(s * W_hh_l0[2 * r0 + 1],  s * W_hh_l0[2 * r1 + 1]);
        bb0[p]   = make2(s * (b_ih_l0[r0] + b_hh_l0[r0]), s * (b_ih_l0[r1] + b_hh_l0[r1]));
        wih1a[p] = make2(s * W_ih_l1[2 * r0 + 0],  s * W_ih_l1[2 * r1 + 0]);
        wih1b[p] = make2(s * W_ih_l1[2 * r0 + 1],  s * W_ih_l1[2 * r1 + 1]);
        whh1a[p] = make2(s * W_hh_l1[2 * r0 + 0],  s * W_hh_l1[2 * r1 + 0]);
        whh1b[p] = make2(s * W_hh_l1[2 * r0 + 1],  s * W_hh_l1[2 * r1 + 1]);
        bb1[p]   = make2(s * (b_ih_l1[r0] + b_hh_l1[r0]), s * (b_ih_l1[r1] + b_hh_l1[r1]));
    }
    const float wd0 = W_dense[0], wd1 = W_dense[1], bd = b_dense[0];

    v2f h0 = splat2(0.f), c0 = splat2(0.f), h1 = splat2(0.f), c1 = splat2(0.f);

#if HAVE_ASYNC_DEV
    if constexpr (STAGE) {
        // Launcher guarantees B%32==0 and T%LSTM_CH==0 for this variant.
        __shared__ float sx[2][32 * LDS_PITCH];   // 9216 B, wave-private block

        // Stage one chunk: 32 rows x 32 timesteps (4 KB) with 8 async b128 per lane.
        auto stage_chunk = [&](int t0, int bi) {
#pragma unroll
            for (int i = 0; i < 8; ++i) {
                const int q = i * 32 + lane;
                const int row = q >> 3, c4 = q & 7;
                const float* g = x + (size_t)(blockIdx.x * 32 + row) * (size_t)T
                                   + t0 + c4 * 4;
                const float* l = &sx[bi][row * LDS_PITCH + c4 * 4];
                async_ld16(g, l);
            }
        };

        const int nch = T / LSTM_CH;
        stage_chunk(0, 0);
        for (int k = 0; k < nch; ++k) {
            if (k + 1 < nch) {
                stage_chunk((k + 1) * LSTM_CH, (k + 1) & 1);   // overlap next copy
                __builtin_amdgcn_s_wait_asynccnt(8);           // chunk k complete
            } else {
                __builtin_amdgcn_s_wait_asynccnt(0);
            }
            const float* row = &sx[k & 1][lane * LDS_PITCH];
            for (int tt = 0; tt < LSTM_CH; tt += 4) {
                const float4 xv = *(const float4*)(row + tt);  // ds_load_b128
                lstm2_step(xv.x, wih0, whh0a, whh0b, bb0, wih1a, wih1b, whh1a, whh1b, bb1, h0, c0, h1, c1);
                lstm2_step(xv.y, wih0, whh0a, whh0b, bb0, wih1a, wih1b, whh1a, whh1b, bb1, h0, c0, h1, c1);
                lstm2_step(xv.z, wih0, whh0a, whh0b, bb0, wih1a, wih1b, whh1a, whh1b, bb1, h0, c0, h1, c1);
                lstm2_step(xv.w, wih0, whh0a, whh0b, bb0, wih1a, wih1b, whh1a, whh1b, bb1, h0, c0, h1, c1);
            }
        }

#if HAVE_WMMA_F32X4
        // ---- dense head on the matrix unit: Out[32] = H[32x2] @ Wd[2x1] + bd.
        // V_WMMA_F32_16X16X4_F32, A: 16x4 f32 (VGPR0=K0, VGPR1=K1 in lanes 0-15;
        // lanes 16-31 feed K2,K3 -> annihilated by zeroing those rows of B).
        // B: 4x16 with every column = (wd0, wd1) for K0,K1. C = bias splat.
        const float hxs = __shfl_xor(h1.x, 16, 32);   // upper<->lower half swap
        const float hys = __shfl_xor(h1.y, 16, 32);
        const bool lo16 = (lane < 16);
        v2f A1; A1.x = h1.x; A1.y = h1.y;            // batches bbase+0..15 as rows
        v2f A2; A2.x = hxs;  A2.y = hys;             // batches bbase+16..31 as rows
        v2f Bm; Bm.x = lo16 ? wd0 : 0.0f; Bm.y = lo16 ? wd1 : 0.0f;
        v8f C;
#pragma unroll
        for (int j = 0; j < 8; ++j) C[j] = bd;
        v8f D1 = __builtin_amdgcn_wmma_f32_16x16x4_f32(
            false, A1, false, Bm, (short)0, C, false, false);
        v8f D2 = __builtin_amdgcn_wmma_f32_16x16x4_f32(
            false, A2, false, Bm, (short)0, C, false, false);
        // D[M][N]: lane==0 holds M=0..7 (N=0), lane==16 holds M=8..15 (N=0).
        const int bbase = blockIdx.x * 32;
        if (lane == 0 || lane == 16) {
            const int off = (lane == 0) ? 0 : 8;
            float4 r0, r1, r2, r3;
            r0.x = fast_sig(D1[0]); r0.y = fast_sig(D1[1]); r0.z = fast_sig(D1[2]); r0.w = fast_sig(D1[3]);
            r1.x = fast_sig(D1[4]); r1.y = fast_sig(D1[5]); r1.z = fast_sig(D1[6]); r1.w = fast_sig(D1[7]);
            r2.x = fast_sig(D2[0]); r2.y = fast_sig(D2[1]); r2.z = fast_sig(D2[2]); r2.w = fast_sig(D2[3]);
            r3.x = fast_sig(D2[4]); r3.y = fast_sig(D2[5]); r3.z = fast_sig(D2[6]); r3.w = fast_sig(D2[7]);
            *(float4*)(out + bbase + off + 0)       = r0;
            *(float4*)(out + bbase + off + 4)       = r1;
            *(float4*)(out + bbase + 16 + off + 0)  = r2;
            *(float4*)(out + bbase + 16 + off + 4)  = r3;
        }
#else
        const float z = __builtin_fmaf(h1.x, wd0, __builtin_fmaf(h1.y, wd1, bd));
        out[b] = fast_sig(z);
#endif
        return;
    }
#endif
    // ---- direct-global fallback path ----
    if (b >= B) return;
    const float* __restrict__ xb = x + (size_t)b * (size_t)T;
    int t = 0;
    for (; t + 4 <= T; t += 4) {
        const float4 xv = *(const float4*)(xb + t);
        __builtin_prefetch(xb + t + 128, 0, 0);   // global_prefetch_b8, ~512B ahead
        lstm2_step(xv.x, wih0, whh0a, whh0b, bb0, wih1a, wih1b, whh1a, whh1b, bb1, h0, c0, h1, c1);
        lstm2_step(xv.y, wih0, whh0a, whh0b, bb0, wih1a, wih1b, whh1a, whh1b, bb1, h0, c0, h1, c1);
        lstm2_step(xv.z, wih0, whh0a, whh0b, bb0, wih1a, wih1b, whh1a, whh1b, bb1, h0, c0, h1, c1);
        lstm2_step(xv.w, wih0, whh0a, whh0b, bb0, wih1a, wih1b, whh1a, whh1b, bb1, h0, c0, h1, c1);
    }
    for (; t < T; ++t) {
        lstm2_step(xb[t], wih0, whh0a, whh0b, bb0, wih1a, wih1b, whh1a, whh1b, bb1, h0, c0, h1, c1);
    }
    const float z = __builtin_fmaf(h1.x, wd0, __builtin_fmaf(h1.y, wd1, bd));
    out[b] = fast_sig(z);
}

// ---------------- launcher ----------------
extern "C" void kernel_launch(void* const* d_in, const int* in_sizes, int n_in,
                              void* d_out, int out_size, void* d_ws, size_t ws_size,
                              hipStream_t stream)
{
    const float* x       = (const float*)d_in[0];
    const float* W_ih_l0 = (const float*)d_in[1];
    const float* W_hh_l0 = (const float*)d_in[2];
    const float* b_ih_l0 = (const float*)d_in[3];
    const float* b_hh_l0 = (const float*)d_in[4];
    const float* W_ih_l1 = (const float*)d_in[5];
    const float* W_hh_l1 = (const float*)d_in[6];
    const float* b_ih_l1 = (const float*)d_in[7];
    const float* b_hh_l1 = (const float*)d_in[8];
    const float* W_dense = (const float*)d_in[9];
    const float* b_dense = (const float*)d_in[10];

    const int B = out_size;            // 16384
    const int T = in_sizes[0] / B;     // 2048
    const int grid = (B + 31) / 32;    // 512 one-wave workgroups -> max WGP spread

    if ((B % 32 == 0) && (T % LSTM_CH == 0) && T >= LSTM_CH) {
        RNN_45561013076393_kernel<true><<<grid, 32, 0, stream>>>(
            x, W_ih_l0, W_hh_l0, b_ih_l0, b_hh_l0,
            W_ih_l1, W_hh_l1, b_ih_l1, b_hh_l1,
            W_dense, b_dense, (float*)d_out, B, T);
    } else {
        RNN_45561013076393_kernel<false><<<grid, 32, 0, stream>>>(
            x, W_ih_l0, W_hh_l0, b_ih_l0, b_hh_l0,
            W_ih_l1, W_hh_l1, b_ih_l1, b_hh_l1,
            W_dense, b_dense, (float*)d_out, B, T);
    }
}